// AdaptLoop_4148938408293
// MI455X (gfx1250) — compile-verified
//
#include <hip/hip_runtime.h>
#include <stdint.h>

#ifndef __has_builtin
#define __has_builtin(x) 0
#endif

#if __has_builtin(__builtin_amdgcn_global_load_async_to_lds_b128)
#define USE_ASYNC_BUILTIN 1
#else
#define USE_ASYNC_BUILTIN 0
#endif

#define T_LEN      32000
#define NCH_TOTAL  1600          // B*F = 32*50
#define CH_PER_WG  32            // one wave32, lane == channel
#define TILE_T     128           // samples staged per tile
#define NTILES     (T_LEN / TILE_T)
#define LDS_STRIDE 132           // floats; 16B-aligned rows, padded vs bank conflicts
#define NLOOP      5

typedef int   v4i __attribute__((vector_size(4 * sizeof(int))));
typedef float v4f __attribute__((ext_vector_type(4)));
typedef __attribute__((address_space(1))) v4i g_v4i;   // global
typedef __attribute__((address_space(3))) v4i l_v4i;   // LDS
typedef __attribute__((address_space(3))) void lvoid;

// ---- async global -> LDS copy primitive (CDNA5 ASYNCcnt path) ---------------
__device__ __forceinline__ void async_load_b128(const float* g, float* l) {
#if USE_ASYNC_BUILTIN
  __builtin_amdgcn_global_load_async_to_lds_b128((g_v4i*)g, (l_v4i*)l, 0, 0);
#else
  uint32_t loff  = (uint32_t)(uintptr_t)(lvoid*)l;   // LDS byte address
  uint64_t gaddr = (uint64_t)(uintptr_t)g;
  asm volatile("global_load_async_to_lds_b128 %0, %1, off"
               :: "v"(loff), "v"(gaddr) : "memory");
#endif
}

__device__ __forceinline__ void wait_async_0() {
  asm volatile("s_wait_asynccnt 0x0" ::: "memory");
}

// -----------------------------------------------------------------------------
__global__ void __launch_bounds__(CH_PER_WG, 1)
adapt_loop_kernel(const float* __restrict__ x,
                  const float* __restrict__ a1,
                  const float* __restrict__ b0,
                  const float* __restrict__ init_state,
                  const float* __restrict__ factor,
                  const float* __restrict__ expfac,
                  const float* __restrict__ offset,
                  const float* __restrict__ corr_p,
                  const float* __restrict__ mult_p,
                  const float* __restrict__ minlvl_p,
                  float* __restrict__ out) {
  __shared__ __align__(16) float s_in[2][CH_PER_WG][LDS_STRIDE];

  const int lane  = threadIdx.x;             // 0..31
  const int cbase = blockIdx.x * CH_PER_WG;  // 50 blocks
  const int ch    = cbase + lane;            // this lane's channel

  // ---- uniform parameters, folded for the fast path -------------------------
  const float LOG2E = 1.4426950408889634f;
  float va1[NLOOP], vb0[NLOOP], vfac[NLOOP], vef2[NLOOP], voff[NLOOP];
  float st[NLOOP], rst[NLOOP];
#pragma unroll
  for (int k = 0; k < NLOOP; ++k) {
    va1[k]  = a1[k];
    vb0[k]  = b0[k];
    vfac[k] = factor[k];
    vef2[k] = expfac[k] * LOG2E;   // exp(expfac*(t-1)) == exp2(vef2*t - vef2)
    voff[k] = offset[k];
    st[k]   = init_state[k];
    rst[k]  = __builtin_amdgcn_rcpf(st[k]);   // 1/state, kept off critical path
  }
  const float corr   = corr_p[0];
  const float mult   = mult_p[0];
  const float minlvl = minlvl_p[0];

  const float* gin_base = x + (size_t)cbase * T_LEN + (size_t)lane * 4;
  float*       gout     = out + (size_t)ch * T_LEN;   // this lane's output row

  // One full adaptation-loop step for one sample (sequential dependency).
  auto step = [&](float v) -> float {
    float tmp = fmaxf(v, minlvl);
#pragma unroll
    for (int k = 0; k < NLOOP; ++k) {
      tmp *= rst[k];                                       // tmp / state[k]
      // sigmoid overshoot limiter: factor/(1+exp(expfac*(tmp-1))) - offset
      float arg = __builtin_fmaf(tmp, vef2[k], -vef2[k]);  // <= 0 when tmp>1
      float e   = __builtin_amdgcn_exp2f(arg);
      float lim = __builtin_fmaf(vfac[k],
                                 __builtin_amdgcn_rcpf(1.0f + e), -voff[k]);
      tmp = (tmp > 1.0f) ? lim : tmp;
      // RC lowpass state update (off the critical path of stage k+1)
      float ns = __builtin_fmaf(va1[k], st[k], vb0[k] * tmp);
      st[k]  = ns;
      rst[k] = __builtin_amdgcn_rcpf(ns);                  // for next timestep
    }
    return (tmp - corr) * mult;
  };

  // ---- prime the pipeline: async-load tile 0 --------------------------------
#pragma unroll
  for (int c = 0; c < CH_PER_WG; ++c)
    async_load_b128(gin_base + (size_t)c * T_LEN, &s_in[0][c][lane * 4]);

  for (int tile = 0; tile < NTILES; ++tile) {
    const int buf = tile & 1;
    const int t0  = tile * TILE_T;

    // Only the prefetch is outstanding here, and it has been in flight for an
    // entire compute phase -> this wait is (almost) free.
    wait_async_0();

    // Prefetch next tile into the other buffer while we compute this one.
    if (tile + 1 < NTILES) {
      const float* gn = gin_base + (size_t)(t0 + TILE_T);
#pragma unroll
      for (int c = 0; c < CH_PER_WG; ++c)
        async_load_b128(gn + (size_t)c * T_LEN, &s_in[buf ^ 1][c][lane * 4]);
    }

    // ---- sequential adaptation loops over this tile -------------------------
    // 4 samples per chunk: one ds_load_b128 in, one global_store_b128 out
    // (STOREcnt, fire-and-forget; no wait on the critical path).
    for (int tt = 0; tt < TILE_T; tt += 4) {
      v4f r;
      r.x = step(s_in[buf][lane][tt + 0]);
      r.y = step(s_in[buf][lane][tt + 1]);
      r.z = step(s_in[buf][lane][tt + 2]);
      r.w = step(s_in[buf][lane][tt + 3]);
      *(v4f*)(gout + t0 + tt) = r;
    }
  }
  // S_ENDPGM performs an implicit wait-idle; STOREcnt drains automatically.
}

// -----------------------------------------------------------------------------
extern "C" void kernel_launch(void* const* d_in, const int* in_sizes, int n_in,
                              void* d_out, int out_size, void* d_ws, size_t ws_size,
                              hipStream_t stream) {
  const float* x          = (const float*)d_in[0];
  const float* a1         = (const float*)d_in[1];
  const float* b0         = (const float*)d_in[2];
  const float* init_state = (const float*)d_in[3];
  const float* factor     = (const float*)d_in[4];
  const float* expfac     = (const float*)d_in[5];
  const float* offset     = (const float*)d_in[6];
  const float* corr       = (const float*)d_in[7];
  const float* mult       = (const float*)d_in[8];
  const float* minlvl     = (const float*)d_in[9];
  float* out = (float*)d_out;

  dim3 grid(NCH_TOTAL / CH_PER_WG);   // 50 workgroups
  dim3 block(CH_PER_WG);              // 32 threads = one wave32
  adapt_loop_kernel<<<grid, block, 0, stream>>>(
      x, a1, b0, init_state, factor, expfac, offset, corr, mult, minlvl, out);
}